// DynamicKRouter_2190433320993
// MI455X (gfx1250) — compile-verified
//
#include <hip/hip_runtime.h>
#include <math.h>

#define BATCH   16384
#define C_CH    64
#define FEAT    1024
#define HID     2048
#define NE      12
#define LN_EPS  1e-5f

typedef __attribute__((ext_vector_type(2))) float v2f;
typedef __attribute__((ext_vector_type(8))) float v8f;

// ---------------------------------------------------------------------------
// Kernel 1: 4x4 adaptive average pool (exact block means, H=W=16 -> 4x4).
// One thread per pooled output element. 4 coalesced float4 loads per thread.
// feat [B, C, 16, 16] -> x [B, C*16] with layout c*16 + sy*4 + sx.
// ---------------------------------------------------------------------------
__global__ __launch_bounds__(256) void pool_kernel(const float* __restrict__ feat,
                                                   float* __restrict__ x) {
    int g = blockIdx.x * blockDim.x + threadIdx.x;   // 0 .. B*C*16-1
    int plane = g >> 4;          // b*C + c
    int cell  = g & 15;          // sy*4 + sx
    int sy = cell >> 2, sx = cell & 3;
    const float* p = feat + (size_t)plane * 256 + sy * 64 + sx * 4;
    float s = 0.f;
#pragma unroll
    for (int i = 0; i < 4; ++i) {
        float4 v = *reinterpret_cast<const float4*>(p + i * 16);
        s += v.x + v.y + v.z + v.w;
    }
    x[g] = s * (1.0f / 16.0f);
}

// ---------------------------------------------------------------------------
// Kernel 2: h = x @ W1  (fp32, V_WMMA_F32_16X16X4_F32). Bias b1 folded into
// the epilogue kernel. 256 threads = 8 waves; WG tile 128(M) x 64(N); each
// wave computes 32x32 via a 2x2 grid of 16x16 WMMA accumulators. K staged
// through LDS in 32-wide slabs.
//
// Fragment layouts per CDNA5 ISA 7.12.2 (f32, wave32):
//   A 16x4 : lane&15 = M row; VGPR0 = K {0|2}, VGPR1 = K {1|3} by lane half.
//   B 4x16 : lane&15 = N col; VGPR0 = K {0|2}, VGPR1 = K {1|3} by lane half.
//   C 16x16: lane&15 = N col; VGPR r = M row r (lanes 0-15) / 8+r (16-31).
//
// LDS layouts:
//   As[row][k]            : A fragment = 2 contiguous floats -> ds_load_b64.
//   Bs[kpair][col*2+k&1]  : B K-pair interleaved so a B fragment is also 2
//                           contiguous floats -> ds_load_b64 (no VGPR movs).
//                           Row stride 160 floats (640B) keeps the kb=2 lane
//                           half on banks 32-63 (conflict-free b64 reads).
// ---------------------------------------------------------------------------
#define TM 128
#define TN 64
#define TK 32
#define AS_STRIDE (TK + 4)     // 36 floats: 144B row stride, 16B-aligned stores
#define BS_STRIDE (2 * TN + 32) // 160 floats per k-pair row

__global__ __launch_bounds__(256) void gemm1_kernel(const float* __restrict__ X,
                                                    const float* __restrict__ W1,
                                                    float* __restrict__ Hout) {
    __shared__ float As[TM][AS_STRIDE];        // 18.0 KB
    __shared__ float Bs[TK / 2][BS_STRIDE];    // 10.0 KB

    const int tid  = threadIdx.x;
    const int lane = tid & 31;
    const int wave = tid >> 5;
    const int wm   = wave & 3;            // M sub-tile (4 x 32 rows)
    const int wn   = wave >> 2;           // N sub-tile (2 x 32 cols)
    const int m0   = blockIdx.y * TM;
    const int n0   = blockIdx.x * TN;

    const int frow = lane & 15;           // fragment row/col
    const int kb   = (lane >> 4) * 2;     // K-pair base: 0 or 2

    v8f acc[2][2] = {};

    for (int k0 = 0; k0 < FEAT; k0 += TK) {
        // Stage A tile: 128x32 floats = 1024 float4, 4 per thread.
#pragma unroll
        for (int it = 0; it < 4; ++it) {
            int idx = tid + it * 256;
            int r = idx >> 3, c4 = idx & 7;
            float4 v = *reinterpret_cast<const float4*>(
                X + (size_t)(m0 + r) * FEAT + k0 + c4 * 4);
            *reinterpret_cast<float4*>(&As[r][c4 * 4]) = v;
        }
        // Stage B tile: 32x64 floats, K-pair interleaved scatter (4 scalar
        // LDS stores per float4; staging is 8x colder than fragment reads).
#pragma unroll
        for (int it = 0; it < 2; ++it) {
            int idx = tid + it * 256;
            int r = idx >> 4, c4 = idx & 15;          // r = k row 0..31
            float4 v = *reinterpret_cast<const float4*>(
                W1 + (size_t)(k0 + r) * HID + n0 + c4 * 4);
            int kp = r >> 1, j = r & 1;
            float* bp = &Bs[kp][c4 * 8 + j];          // col*2 + j, col=c4*4..
            bp[0] = v.x; bp[2] = v.y; bp[4] = v.z; bp[6] = v.w;
        }
        __syncthreads();

#pragma unroll
        for (int kk = 0; kk < TK; kk += 4) {
            v2f a[2], b[2];
#pragma unroll
            for (int mi = 0; mi < 2; ++mi) {
                const float* ap = &As[wm * 32 + mi * 16 + frow][kk + kb];
                a[mi] = *reinterpret_cast<const v2f*>(ap);   // ds_load_b64
            }
#pragma unroll
            for (int ni = 0; ni < 2; ++ni) {
                int col = wn * 32 + ni * 16 + frow;
                const float* bp = &Bs[(kk + kb) >> 1][col * 2];
                b[ni] = *reinterpret_cast<const v2f*>(bp);   // ds_load_b64
            }
#pragma unroll
            for (int mi = 0; mi < 2; ++mi)
#pragma unroll
                for (int ni = 0; ni < 2; ++ni)
                    acc[mi][ni] = __builtin_amdgcn_wmma_f32_16x16x4_f32(
                        false, a[mi], false, b[ni],
                        (short)0, acc[mi][ni], false, false);
        }
        __syncthreads();
    }

    // Store C per documented 16x16 f32 layout.
    const int colC  = lane & 15;
    const int rbase = (lane >> 4) * 8;
#pragma unroll
    for (int mi = 0; mi < 2; ++mi)
#pragma unroll
        for (int ni = 0; ni < 2; ++ni) {
            size_t base = (size_t)(m0 + wm * 32 + mi * 16 + rbase) * HID
                        + n0 + wn * 32 + ni * 16 + colC;
#pragma unroll
            for (int r = 0; r < 8; ++r)
                Hout[base + (size_t)r * HID] = acc[mi][ni][r];
        }
}

// ---------------------------------------------------------------------------
// Kernel 3: per-row epilogue. One wave32 per batch row (8 rows per 256-thread
// block). +b1, LayerNorm (two-pass, exact), erf-GELU, h@W2+b2, softmax,
// dynamic-k top-4 routing.
// ---------------------------------------------------------------------------
__global__ __launch_bounds__(256) void route_kernel(
    const float* __restrict__ Hin,   const float* __restrict__ b1,
    const float* __restrict__ gamma, const float* __restrict__ beta,
    const float* __restrict__ W2,    const float* __restrict__ b2,
    const float* __restrict__ log_threshold,
    float* __restrict__ gate_out, float* __restrict__ k_out,
    float* __restrict__ loss_out) {
    const int lane = threadIdx.x & 31;
    const int wave = threadIdx.x >> 5;
    const int row  = blockIdx.x * 8 + wave;
    const float* hrow = Hin + (size_t)row * HID;

    // Load the row (+bias) into registers: 16 float4 = 64 floats per lane.
    float4 hv[16];
    float s = 0.f;
#pragma unroll
    for (int t = 0; t < 16; ++t) {
        int v4 = lane + 32 * t;
        float4 h4 = *reinterpret_cast<const float4*>(hrow + (size_t)v4 * 4);
        float4 bb = *reinterpret_cast<const float4*>(b1 + v4 * 4);
        h4.x += bb.x; h4.y += bb.y; h4.z += bb.z; h4.w += bb.w;
        hv[t] = h4;
        s += h4.x + h4.y + h4.z + h4.w;
    }
#pragma unroll
    for (int off = 16; off > 0; off >>= 1) s += __shfl_xor(s, off, 32);
    const float mu = s * (1.0f / HID);

    float vs = 0.f;
#pragma unroll
    for (int t = 0; t < 16; ++t) {
        float dx = hv[t].x - mu; vs += dx * dx;
        dx = hv[t].y - mu; vs += dx * dx;
        dx = hv[t].z - mu; vs += dx * dx;
        dx = hv[t].w - mu; vs += dx * dx;
    }
#pragma unroll
    for (int off = 16; off > 0; off >>= 1) vs += __shfl_xor(vs, off, 32);
    const float rstd = rsqrtf(vs * (1.0f / HID) + LN_EPS);

    // LN + erf-GELU + partial dot with W2 (rows are 12 floats = 3 float4).
    float acc[NE];
#pragma unroll
    for (int j = 0; j < NE; ++j) acc[j] = 0.f;
    const float inv_sqrt2 = 0.70710678118654752440f;
#pragma unroll
    for (int t = 0; t < 16; ++t) {
        int e0 = (lane + 32 * t) * 4;
        float4 gg = *reinterpret_cast<const float4*>(gamma + e0);
        float4 be = *reinterpret_cast<const float4*>(beta + e0);
        float hval[4] = {hv[t].x, hv[t].y, hv[t].z, hv[t].w};
        float gval[4] = {gg.x, gg.y, gg.z, gg.w};
        float bval[4] = {be.x, be.y, be.z, be.w};
#pragma unroll
        for (int c = 0; c < 4; ++c) {
            float xn = (hval[c] - mu) * rstd * gval[c] + bval[c];
            float ge = 0.5f * xn * (1.0f + erff(xn * inv_sqrt2));
            const float* wr = W2 + (size_t)(e0 + c) * NE;
            float4 w0 = *reinterpret_cast<const float4*>(wr);
            float4 w1 = *reinterpret_cast<const float4*>(wr + 4);
            float4 w2 = *reinterpret_cast<const float4*>(wr + 8);
            acc[0]  += ge * w0.x; acc[1]  += ge * w0.y;
            acc[2]  += ge * w0.z; acc[3]  += ge * w0.w;
            acc[4]  += ge * w1.x; acc[5]  += ge * w1.y;
            acc[6]  += ge * w1.z; acc[7]  += ge * w1.w;
            acc[8]  += ge * w2.x; acc[9]  += ge * w2.y;
            acc[10] += ge * w2.z; acc[11] += ge * w2.w;
        }
    }
#pragma unroll
    for (int j = 0; j < NE; ++j)
#pragma unroll
        for (int off = 16; off > 0; off >>= 1)
            acc[j] += __shfl_xor(acc[j], off, 32);

    // All lanes now hold the full 12 logits; finish routing redundantly.
    float logits[NE];
#pragma unroll
    for (int j = 0; j < NE; ++j) logits[j] = acc[j] + b2[j];

    float mx = logits[0];
#pragma unroll
    for (int j = 1; j < NE; ++j) mx = fmaxf(mx, logits[j]);
    float p[NE]; float psum = 0.f;
#pragma unroll
    for (int j = 0; j < NE; ++j) { p[j] = __expf(logits[j] - mx); psum += p[j]; }
    float pinv = 1.0f / psum;
#pragma unroll
    for (int j = 0; j < NE; ++j) p[j] *= pinv;

    const float lt  = log_threshold[0];
    const float thr = fmaxf(1.0f / (1.0f + expf(-lt)), 1.0f / NE);
    int cnt = 0;
#pragma unroll
    for (int j = 0; j < NE; ++j) cnt += (p[j] > thr) ? 1 : 0;
    int active_k = cnt < 1 ? 1 : (cnt > 4 ? 4 : cnt);

    // Top-4 (lowest index wins ties, matching lax.top_k).
    float q[NE], gw[NE];
#pragma unroll
    for (int j = 0; j < NE; ++j) { q[j] = p[j]; gw[j] = 0.f; }
    float gsum = 0.f;
#pragma unroll
    for (int k = 0; k < 4; ++k) {
        float best = -1.f; int bi = 0;
#pragma unroll
        for (int j = 0; j < NE; ++j)
            if (q[j] > best) { best = q[j]; bi = j; }
#pragma unroll
        for (int j = 0; j < NE; ++j)
            if (j == bi) {
                q[j] = -2.f;
                if (k < active_k) { gw[j] = best; gsum += best; }
            }
    }
    const float ginv = 1.0f / (gsum + 1e-8f);
    if (lane < NE) gate_out[(size_t)row * NE + lane] = gw[lane] * ginv;
    if (lane == 0) k_out[row] = (float)active_k;
    if (row == 0 && lane == 0) loss_out[0] = 0.0f;
}

// ---------------------------------------------------------------------------
extern "C" void kernel_launch(void* const* d_in, const int* in_sizes, int n_in,
                              void* d_out, int out_size, void* d_ws, size_t ws_size,
                              hipStream_t stream) {
    const float* feat  = (const float*)d_in[0];
    const float* W1    = (const float*)d_in[1];
    const float* b1    = (const float*)d_in[2];
    const float* gamma = (const float*)d_in[3];
    const float* beta  = (const float*)d_in[4];
    const float* W2    = (const float*)d_in[5];
    const float* b2    = (const float*)d_in[6];
    const float* logth = (const float*)d_in[7];

    float* out      = (float*)d_out;
    float* gate_out = out;                          // [B, 12]
    float* k_out    = out + (size_t)BATCH * NE;     // [B]
    float* loss_out = k_out + BATCH;                // [1]

    float* x = (float*)d_ws;                        // [B, 1024]  (64 MB)
    float* h = x + (size_t)BATCH * FEAT;            // [B, 2048]  (128 MB)

    // 1) pooling: B*C*16 threads
    {
        int total = BATCH * C_CH * 16;
        pool_kernel<<<total / 256, 256, 0, stream>>>(feat, x);
    }
    // 2) GEMM1: grid (N/64, M/128)
    {
        dim3 grid(HID / TN, BATCH / TM);
        gemm1_kernel<<<grid, 256, 0, stream>>>(x, W1, h);
    }
    // 3) routing epilogue: one wave per row, 8 rows per block
    {
        route_kernel<<<BATCH / 8, 256, 0, stream>>>(h, b1, gamma, beta, W2, b2,
                                                    logth, gate_out, k_out, loss_out);
    }
}